// HybridQuantizer_2345052144228
// MI455X (gfx1250) — compile-verified
//
#include <hip/hip_runtime.h>
#include <hip/hip_bf16.h>
#include <stdint.h>

// Problem constants (from reference): x:[N,8192] f32, W:[512,8192] f32,
// out[n,:] = W[:, argmax(x[n,:])]  -> argmax + gather, memory bound.
#define QUANT_DIM 8192
#define OUT_DIM   512

// ---------------------------------------------------------------------------
// TDM (Tensor Data Mover) support. This toolchain (clang-23 / therock-10.0)
// declares the 6-arg builtin:
//   (uint32x4 g0, int32x8 g1, int32x4 g2, int32x4 g3, int32x8 g4, i32 cpol)
// Guarded so the host pass / other toolchains fall back to cooperative copies.
// ---------------------------------------------------------------------------
#if defined(__HIP_DEVICE_COMPILE__) && defined(__gfx1250__) && \
    __has_builtin(__builtin_amdgcn_tensor_load_to_lds) &&       \
    __has_builtin(__builtin_amdgcn_s_wait_tensorcnt)
#define USE_TDM 1
#else
#define USE_TDM 0
#endif

#if USE_TDM
typedef unsigned int u32x4 __attribute__((ext_vector_type(4)));
typedef int          i32x8 __attribute__((ext_vector_type(8)));
typedef int          i32x4 __attribute__((ext_vector_type(4)));

// Build a D# descriptor (CDNA5 ISA) for a 1-D row load:
// data_size=4B, tensor_dim0=tile_dim0=QUANT_DIM, tensor_dim1=tile_dim1=1.
__device__ __forceinline__ void tdm_load_row_to_lds(const float* gsrc,
                                                    unsigned lds_off) {
  const uint64_t ga = (uint64_t)(uintptr_t)gsrc;
  u32x4 g0;
  // group0: count=1 (valid user descriptor), lds_addr, global_addr, type=2
  g0[0] = 1u;                                            // count=1, no gather
  g0[1] = lds_off;                                       // lds_addr [63:32]
  g0[2] = (unsigned)(ga & 0xFFFFFFFFu);                  // global_addr[31:0]
  g0[3] = (unsigned)((ga >> 32) & 0x01FFFFFFu)           // global_addr[56:32]
          | 0x80000000u;                                 // type=2 ("image")
  i32x8 g1;
  g1[0] = 0x00020000;                    // data_size=2 (4 bytes); no mask/pad
  g1[1] = (int)((QUANT_DIM & 0xFFFF) << 16);   // tensor_dim0[15:0] @ [31:16]
  g1[2] = (int)(((QUANT_DIM >> 16) & 0xFFFF)   // tensor_dim0[31:16]
          | (1u << 16));                       // tensor_dim1 = 1
  g1[3] = (int)((QUANT_DIM & 0xFFFF) << 16);   // tile_dim0 = 8192 @ [127:112]
  g1[4] = 1;                                   // tile_dim1 = 1, tile_dim2 = 0
  g1[5] = QUANT_DIM;                           // tensor_dim0_stride[31:0]
  g1[6] = 0;                                   // stride0[47:32], stride1[15:0]
  g1[7] = 0;                                   // stride1[47:16]
  i32x4 g2 = {0, 0, 0, 0};
  i32x4 g3 = {0, 0, 0, 0};
  i32x8 g4 = {0, 0, 0, 0, 0, 0, 0, 0};         // unused for <=2D tensors
  __builtin_amdgcn_tensor_load_to_lds(g0, g1, g2, g3, g4, 0);
}
#endif

// ---------------------------------------------------------------------------
// Kernel 1: W[512,8192] -> WT[8192,512] tiled transpose (both sides coalesced)
// ---------------------------------------------------------------------------
__global__ void __launch_bounds__(256)
hq_transpose_kernel(const float* __restrict__ W, float* __restrict__ WT) {
  __shared__ float tile[32][33];          // +1 to kill bank conflicts
  const int cb = blockIdx.x * 32;         // column base in [0, QUANT_DIM)
  const int rb = blockIdx.y * 32;         // row base    in [0, OUT_DIM)
  const int tx = threadIdx.x;             // 0..31
  const int ty = threadIdx.y;             // 0..7
#pragma unroll
  for (int k = 0; k < 32; k += 8)
    tile[ty + k][tx] = W[(size_t)(rb + ty + k) * QUANT_DIM + cb + tx];
  __syncthreads();
#pragma unroll
  for (int k = 0; k < 32; k += 8)
    WT[(size_t)(cb + ty + k) * OUT_DIM + rb + tx] = tile[tx][ty + k];
}

// ---------------------------------------------------------------------------
// Kernel 2: per-row argmax (TDM -> LDS -> reduce) + gather of the W column.
// WT_LAYOUT=true : Wsrc = WT[8192,512], gather row is contiguous (fast path)
// WT_LAYOUT=false: Wsrc = W[512,8192],  strided gather (L2-resident fallback)
// ---------------------------------------------------------------------------
template <bool WT_LAYOUT>
__global__ void __launch_bounds__(256)
hq_argmax_gather_kernel(const float* __restrict__ x,
                        const float* __restrict__ Wsrc,
                        float* __restrict__ out) {
  __shared__ __align__(16) float tile[QUANT_DIM];   // 32 KB row buffer
  __shared__ float wval[8];
  __shared__ int   widx[8];
  __shared__ int   sFinal;

  const int row = blockIdx.x;
  const float* src = x + (size_t)row * QUANT_DIM;

#if USE_TDM
  // Wave 0 issues one TDM descriptor for the whole 32 KB row, then waits on
  // TENSORcnt; the other 7 waves park at the barrier. Cross-block overlap
  // (~9 blocks/WGP at 32KB LDS each) hides the DMA latency.
  if (threadIdx.x < 32) {
    tdm_load_row_to_lds(src, (unsigned)(uintptr_t)&tile[0]);
    __builtin_amdgcn_s_wait_tensorcnt(0);
  }
#else
  {
    const float4* s4 = (const float4*)src;
    float4* t4 = (float4*)tile;
    for (int i = threadIdx.x; i < QUANT_DIM / 4; i += 256) t4[i] = s4[i];
  }
#endif
  __syncthreads();

  // Thread-local scan: 8 x b128 LDS loads, ascending indices so strict '>'
  // keeps the FIRST occurrence of the max (jnp.argmax semantics).
  float best = -3.402823466e+38f;
  int   bidx = 0;
  const float4* t4 = (const float4*)tile;
#pragma unroll
  for (int k = 0; k < 8; ++k) {
    const int vi = threadIdx.x + k * 256;
    const float4 v = t4[vi];
    const int base = vi * 4;
    if (v.x > best) { best = v.x; bidx = base; }
    if (v.y > best) { best = v.y; bidx = base + 1; }
    if (v.z > best) { best = v.z; bidx = base + 2; }
    if (v.w > best) { best = v.w; bidx = base + 3; }
  }

  // wave32 butterfly reduce; tie-break on smaller index.
#pragma unroll
  for (int off = 16; off > 0; off >>= 1) {
    const float ov = __shfl_xor(best, off, 32);
    const int   oi = __shfl_xor(bidx, off, 32);
    if (ov > best || (ov == best && oi < bidx)) { best = ov; bidx = oi; }
  }
  const int lane = threadIdx.x & 31;
  const int wid  = threadIdx.x >> 5;
  if (lane == 0) { wval[wid] = best; widx[wid] = bidx; }
  __syncthreads();
  if (threadIdx.x == 0) {
    float b = wval[0];
    int  bi = widx[0];
#pragma unroll
    for (int w = 1; w < 8; ++w)
      if (wval[w] > b || (wval[w] == b && widx[w] < bi)) { b = wval[w]; bi = widx[w]; }
    sFinal = bi;
  }
  __syncthreads();
  const int idx = sFinal;

  if (WT_LAYOUT) {
    // Contiguous 2 KB row of WT: fully coalesced float4 copy.
    if (threadIdx.x < OUT_DIM / 4) {
      const float4* srow = (const float4*)(Wsrc + (size_t)idx * OUT_DIM);
      float4* drow = (float4*)(out + (size_t)row * OUT_DIM);
      drow[threadIdx.x] = srow[threadIdx.x];
    }
  } else {
    // Strided column gather of W (32 KB stride); W (16 MB) lives in L2.
#pragma unroll
    for (int j = threadIdx.x; j < OUT_DIM; j += 256)
      out[(size_t)row * OUT_DIM + j] = Wsrc[(size_t)j * QUANT_DIM + idx];
  }
}

// ---------------------------------------------------------------------------
extern "C" void kernel_launch(void* const* d_in, const int* in_sizes, int n_in,
                              void* d_out, int out_size, void* d_ws, size_t ws_size,
                              hipStream_t stream) {
  const float* x = (const float*)d_in[0];   // [N, QUANT_DIM] f32
  const float* W = (const float*)d_in[1];   // [OUT_DIM, QUANT_DIM] f32
  float* out = (float*)d_out;               // [N, OUT_DIM] f32
  const int N = in_sizes[0] / QUANT_DIM;

  const size_t WT_BYTES = (size_t)OUT_DIM * QUANT_DIM * sizeof(float); // 16 MB
  if (ws_size >= WT_BYTES) {
    float* WT = (float*)d_ws;
    dim3 tg(QUANT_DIM / 32, OUT_DIM / 32);  // 256 x 16 blocks
    hq_transpose_kernel<<<tg, dim3(32, 8), 0, stream>>>(W, WT);
    hq_argmax_gather_kernel<true><<<N, 256, 0, stream>>>(x, WT, out);
  } else {
    hq_argmax_gather_kernel<false><<<N, 256, 0, stream>>>(x, W, out);
  }
}